// PPM_54657753808985
// MI455X (gfx1250) — compile-verified
//
#include <hip/hip_runtime.h>
#include <hip/hip_bf16.h>

typedef __attribute__((ext_vector_type(2))) float v2f;
typedef __attribute__((ext_vector_type(8))) float v8f;
typedef int v4i_gcc __attribute__((vector_size(16)));
typedef __attribute__((address_space(1))) v4i_gcc as1_v4i;
typedef __attribute__((address_space(3))) v4i_gcc as3_v4i;

#define B_   32
#define C_   256
#define P_   2304
#define QT   64
#define PT   32
#define NQT  (P_ / QT)   // 36
#define NPT  (P_ / PT)   // 72
#define EPS_ 1e-8f

#if defined(__has_builtin)
#if __has_builtin(__builtin_amdgcn_global_load_async_to_lds_b128)
#define ASYNC_LDS 1
#endif
#endif

// 16-byte global -> LDS copy: async DMA (ASYNCcnt) when available.
static __device__ __forceinline__ void async_cp16(const float* gsrc, float* ldst) {
#ifdef ASYNC_LDS
  __builtin_amdgcn_global_load_async_to_lds_b128((as1_v4i*)gsrc, (as3_v4i*)ldst,
                                                 0, 0);
#else
  ldst[0] = gsrc[0]; ldst[1] = gsrc[1]; ldst[2] = gsrc[2]; ldst[3] = gsrc[3];
#endif
}

static __device__ __forceinline__ void wait_async() {
#ifdef ASYNC_LDS
#if __has_builtin(__builtin_amdgcn_s_wait_asynccnt)
  __builtin_amdgcn_s_wait_asynccnt(0);
#else
  asm volatile("s_wait_asynccnt 0x0" ::: "memory");
#endif
#endif
}

static __device__ __forceinline__ v8f wmma_f32(v2f a, v2f b, v8f c) {
  // D = A(16x4 f32) * B(4x16 f32) + C(16x16 f32)
  return __builtin_amdgcn_wmma_f32_16x16x4_f32(false, a, false, b, (short)0, c,
                                               false, false);
}

// ---------------- Kernel 1: reciprocal per-pixel channel norms ----------------
__global__ __launch_bounds__(256) void rnorm_kernel(const float* __restrict__ x,
                                                    float* __restrict__ rnrm) {
  int i = blockIdx.x * 256 + threadIdx.x;        // i = b*P + p
  int b = i / P_, p = i - b * P_;
  const float* xp = x + (size_t)b * C_ * P_ + p;
  float ss = 0.f;
#pragma unroll 8
  for (int c = 0; c < C_; ++c) {
    float t = xp[(size_t)c * P_];
    ss += t * t;
  }
  rnrm[i] = 1.0f / fmaxf(sqrtf(ss), EPS_);
}

// ---------------- Kernel 2: g = Wc @ x + bc  (1x1 conv), WMMA f32 ----------------
__global__ __launch_bounds__(256) void gconv_kernel(const float* __restrict__ x,
                                                    const float* __restrict__ Wc,
                                                    const float* __restrict__ bc,
                                                    float* __restrict__ g) {
  __shared__ float Wl[C_][68];   // Wc[:, k0:k0+64]; stride 68: per-lane-row reads
  __shared__ float xc[64][72];   // x[b, k0:k0+64, p0:p0+64]; stride 72: row-pair reads

  const int tid = threadIdx.x, lane = tid & 31, wave = tid >> 5;
  const int m = lane & 15, hi = lane >> 4;
  const int b = blockIdx.x / (P_ / 64);
  const int p0 = (blockIdx.x - b * (P_ / 64)) * 64;
  const float* xb = x + (size_t)b * C_ * P_;

  v8f acc[2][4];
  for (int ci = 0; ci < 2; ++ci)
    for (int pi = 0; pi < 4; ++pi)
      for (int v = 0; v < 8; ++v) acc[ci][pi][v] = 0.f;

  for (int kc = 0; kc < C_ / 64; ++kc) {
    const int k0 = kc * 64;
    __syncthreads();                                     // tiles free for rewrite
    for (int idx = tid; idx < C_ * 64 / 4; idx += 256) { // Wc chunk (async DMA)
      int c = idx >> 4, k4 = (idx & 15) * 4;
      async_cp16(Wc + (size_t)c * C_ + k0 + k4, &Wl[c][k4]);
    }
    for (int idx = tid; idx < 64 * 64 / 4; idx += 256) { // x chunk (async DMA)
      int kk = idx >> 4, p4 = (idx & 15) * 4;
      async_cp16(xb + (size_t)(k0 + kk) * P_ + p0 + p4, &xc[kk][p4]);
    }
    wait_async();
    __syncthreads();
    for (int ci = 0; ci < 2; ++ci) {
      const int cb = (wave + ci * 8) * 16;
      for (int pi = 0; pi < 4; ++pi) {
        const int pb = pi * 16;
#pragma unroll
        for (int kk = 0; kk < 64; kk += 4) {
          v2f av, bv;
          av.x = Wl[cb + m][kk + 2 * hi];        // A[M=c][K]
          av.y = Wl[cb + m][kk + 2 * hi + 1];
          bv.x = xc[kk + 2 * hi][pb + m];        // B[K][N=p]
          bv.y = xc[kk + 2 * hi + 1][pb + m];
          acc[ci][pi] = wmma_f32(av, bv, acc[ci][pi]);
        }
      }
    }
  }
  for (int ci = 0; ci < 2; ++ci)
    for (int pi = 0; pi < 4; ++pi)
#pragma unroll
      for (int v = 0; v < 8; ++v) {
        int c = (wave + ci * 8) * 16 + v + 8 * hi;   // D: M = vgpr + 8*(lane>=16)
        int p = p0 + pi * 16 + m;
        g[((size_t)b * C_ + c) * P_ + p] = acc[ci][pi][v] + bc[c];
      }
}

// ---------------- Kernel 3: fused  out = g @ relu(xn^T xn)^2 ----------------
// Normalization is folded into the S-tile store (relu(dot*rp*rq)=relu(dot)*rp*rq),
// so every tile is a raw async-DMA copy and fetches pipeline against WMMA phases.
__global__ __launch_bounds__(256) void ppm_kernel(const float* __restrict__ x,
                                                  const float* __restrict__ g,
                                                  const float* __restrict__ rnrm,
                                                  float* __restrict__ out) {
  __shared__ float xp[C_][40];    // raw x p-tile (resident); row-pair reads
  __shared__ float buf[C_][72];   // raw x q-tile;            row-pair reads
  __shared__ float gbuf[C_][68];  // g q-tile;                per-lane-row reads
  __shared__ float sq[PT][68];    // S = (relu(cos))^2;       per-lane-row reads

  const int tid = threadIdx.x, lane = tid & 31, wave = tid >> 5;
  const int m = lane & 15, hi = lane >> 4;
  const int b = blockIdx.x / NPT;
  const int p0 = (blockIdx.x - b * NPT) * PT;
  const float* xb = x + (size_t)b * C_ * P_;
  const float* gb = g + (size_t)b * C_ * P_;
  const float* rb = rnrm + (size_t)b * P_;

  const int spb = (wave >> 2) * 16;   // this wave's S p-subtile
  const int sqb = (wave & 3) * 16;    // this wave's S q-subtile

  // resident raw p-tile + first q-tile (both async)
  for (int idx = tid; idx < C_ * PT / 4; idx += 256) {
    int c = idx >> 3, p4 = (idx & 7) * 4;
    async_cp16(xb + (size_t)c * P_ + p0 + p4, &xp[c][p4]);
  }
  for (int idx = tid; idx < C_ * QT / 4; idx += 256) {
    int c = idx >> 4, q4 = (idx & 15) * 4;
    async_cp16(xb + (size_t)c * P_ + q4, &buf[c][q4]);
  }

  // per-lane p-norm factors for the S store (fixed across q-tiles)
  float rpv[8];
#pragma unroll
  for (int v = 0; v < 8; ++v) rpv[v] = rb[p0 + spb + v + 8 * hi];

  v8f acc[2][2];
  for (int ci = 0; ci < 2; ++ci)
    for (int pi = 0; pi < 2; ++pi)
      for (int v = 0; v < 8; ++v) acc[ci][pi][v] = 0.f;

  for (int qt = 0; qt < NQT; ++qt) {
    const int q0 = qt * QT;
    wait_async();
    __syncthreads();          // xq(qt) ready in buf; gbuf readers of qt-1 done

    // fetch g(qt) while phase A computes on xq(qt)
    for (int idx = tid; idx < C_ * QT / 4; idx += 256) {
      int c = idx >> 4, q4 = (idx & 15) * 4;
      async_cp16(gb + (size_t)c * P_ + q0 + q4, &gbuf[c][q4]);
    }

    {  // Phase A: raw Gram subtile, then fused relu/normalize/square
      v8f s;
      for (int v = 0; v < 8; ++v) s[v] = 0.f;
#pragma unroll 8
      for (int k0 = 0; k0 < C_; k0 += 4) {
        v2f av, bv;
        av.x = xp[k0 + 2 * hi][spb + m];       // A[M=p][K=c]
        av.y = xp[k0 + 2 * hi + 1][spb + m];
        bv.x = buf[k0 + 2 * hi][sqb + m];      // B[K=c][N=q]
        bv.y = buf[k0 + 2 * hi + 1][sqb + m];
        s = wmma_f32(av, bv, s);
      }
      float rq = rb[q0 + sqb + m];
#pragma unroll
      for (int v = 0; v < 8; ++v) {
        float t = fmaxf(s[v], 0.f) * rpv[v] * rq;   // relu(cos), gamma=2
        sq[spb + v + 8 * hi][sqb + m] = t * t;
      }
    }

    wait_async();
    __syncthreads();          // g(qt) in LDS; S tile visible; buf free

    if (qt + 1 < NQT) {       // fetch xq(qt+1) while phase B computes
      for (int idx = tid; idx < C_ * QT / 4; idx += 256) {
        int c = idx >> 4, q4 = (idx & 15) * 4;
        async_cp16(xb + (size_t)c * P_ + q0 + QT + q4, &buf[c][q4]);
      }
    }

    // Phase B: acc[c,p] += g[c, q-tile] . S^T
    for (int ci = 0; ci < 2; ++ci) {
      const int cb = (wave + ci * 8) * 16;
      for (int pi = 0; pi < 2; ++pi) {
        const int pb = pi * 16;
#pragma unroll
        for (int kk = 0; kk < QT; kk += 4) {
          v2f av, bv;
          av.x = gbuf[cb + m][kk + 2 * hi];    // A[M=c][K=q] = g
          av.y = gbuf[cb + m][kk + 2 * hi + 1];
          bv.x = sq[pb + m][kk + 2 * hi];      // B[K=q][N=p] = s[p][q]
          bv.y = sq[pb + m][kk + 2 * hi + 1];
          acc[ci][pi] = wmma_f32(av, bv, acc[ci][pi]);
        }
      }
    }
  }

  for (int ci = 0; ci < 2; ++ci)
    for (int pi = 0; pi < 2; ++pi)
#pragma unroll
      for (int v = 0; v < 8; ++v) {
        int c = (wave + ci * 8) * 16 + v + 8 * hi;
        int p = p0 + pi * 16 + m;
        out[((size_t)b * C_ + c) * P_ + p] = acc[ci][pi][v];
      }
}

extern "C" void kernel_launch(void* const* d_in, const int* in_sizes, int n_in,
                              void* d_out, int out_size, void* d_ws, size_t ws_size,
                              hipStream_t stream) {
  const float* x  = (const float*)d_in[0];
  const float* Wc = (const float*)d_in[1];
  const float* bc = (const float*)d_in[2];
  float* out = (float*)d_out;

  float* g    = (float*)d_ws;                       // B*C*P floats (75.5 MB)
  float* rnrm = g + (size_t)B_ * C_ * P_;           // B*P floats (294 KB)

  rnorm_kernel<<<B_ * P_ / 256, 256, 0, stream>>>(x, rnrm);
  gconv_kernel<<<B_ * (P_ / 64), 256, 0, stream>>>(x, Wc, bc, g);
  ppm_kernel<<<B_ * NPT, 256, 0, stream>>>(x, g, rnrm, out);
}